// LaplaceLoss_gpu_pzy_35570919145579
// MI455X (gfx1250) — compile-verified
//
#include <hip/hip_runtime.h>
#include <math.h>

#define CLASSES 21
#define SPIX    1000
#define SPAD    1008          // S padded to multiple of 16
#define MPAD    32            // 21 class rows padded to 2x16
#define BATCH   8
#define PIX     (512*512)     // 262144 = 1<<18
#define NTILES  (SPAD/16)     // 63

typedef __attribute__((ext_vector_type(2))) float v2f;
typedef __attribute__((ext_vector_type(8))) float v8f;

// ---------------- workspace zeroing ----------------
__global__ void k_zero(float* __restrict__ ws, int n) {
    int i = blockIdx.x * blockDim.x + threadIdx.x;
    if (i < n) ws[i] = 0.0f;
}

// ---------------- segment sum (scrambled reshape, faithful to reference) ----
// pred native flat index tid = b*(C*P) + co*P + hw  (coalesced read)
// HWC flat index q = hw*C + co ; scrambled class c = q/P, pixel p = q%P
__global__ void __launch_bounds__(256)
k_segsum(const float* __restrict__ pred, const int* __restrict__ seg,
         float* __restrict__ R) {
    unsigned int tid = blockIdx.x * blockDim.x + threadIdx.x; // < B*C*P (exact grid)
    unsigned int b   = tid / (CLASSES * PIX);
    unsigned int rem = tid - b * (CLASSES * PIX);
    unsigned int co  = rem >> 18;            // / PIX
    unsigned int hw  = rem & (PIX - 1);
    unsigned int q   = hw * CLASSES + co;    // < 2^23
    unsigned int c   = q >> 18;              // scrambled class, < 21
    unsigned int p   = q & (PIX - 1);
    int s = seg[b * PIX + p];
    float v = pred[tid];
    if (s >= 0 && s < SPIX)
        atomicAdd(&R[((size_t)b * MPAD + c) * SPAD + (unsigned)s], v);
}

// ---------------- valid-id count ----------------
__global__ void k_mask(const int* __restrict__ seg, unsigned int* __restrict__ cnt) {
    unsigned int local = 0;
    int stride = gridDim.x * blockDim.x;
    for (int i = blockIdx.x * blockDim.x + threadIdx.x; i < BATCH * PIX; i += stride) {
        int s = seg[i];
        local += (s >= 0 && s < SPIX) ? 1u : 0u;
    }
    for (int off = 16; off; off >>= 1) local += __shfl_xor(local, off, 32);
    if ((threadIdx.x & 31) == 0) atomicAdd(cnt, local);
}

// ---------------- ||W||_F^2 ----------------
__global__ void k_wnorm(const float* __restrict__ W, float* __restrict__ wss, int n) {
    float local = 0.0f;
    int stride = gridDim.x * blockDim.x;
    for (int i = blockIdx.x * blockDim.x + threadIdx.x; i < n; i += stride) {
        float v = W[i];
        local += v * v;
    }
    for (int off = 16; off; off >>= 1) local += __shfl_xor(local, off, 32);
    if ((threadIdx.x & 31) == 0) atomicAdd(wss, local);
}

// ---------------- trace(R L R^T) via V_WMMA_F32_16X16X4_F32 ----------------
// One wave per (batch, m_tile in {0,1}, n_tile in 0..62).
// A frag (16x4 f32): lane 0-15 -> M=lane, v[0]=K(+0/+2 per half), v[1]=K+1
// B frag (4x16 f32): lane&15 -> N,     v[0]=K(+0/+2 per half), v[1]=K+1
// C/D (16x16 f32):   VGPR v -> M=v (+8 for lanes 16-31), N = lane&15
__global__ void __launch_bounds__(32)
k_trace(const float* __restrict__ L, const float* __restrict__ R,
        float* __restrict__ tsum) {
    const int lane = threadIdx.x;
    const int id   = blockIdx.x;                 // < BATCH*2*NTILES
    const int b    = id / (2 * NTILES);
    const int r    = id - b * (2 * NTILES);
    const int m0   = (r / NTILES) * 16;
    const int n0   = (r % NTILES) * 16;
    const int l15  = lane & 15;
    const int hi   = lane >> 4;                  // 0 or 1

    const float* __restrict__ Rb = R + (size_t)b * MPAD * SPAD;

    const int  t   = n0 + l15;                   // output column, < SPAD
    const bool tin = (t < SPIX);
    const int  tc  = tin ? t : (SPIX - 1);       // clamped: loads always in-bounds

    // B-matrix base for this lane's (clamped) column; k advances by row stride
    const float* __restrict__ bcol = L + (size_t)b * SPIX * SPIX + tc;
    const float* __restrict__ arow = Rb + (size_t)(m0 + l15) * SPAD; // zero-padded rows

    v8f acc = {};
    for (int k0 = 0; k0 < SPIX; k0 += 4) {
        const int ka = k0 + hi * 2;
        v2f a;
        a.x = arow[ka];
        a.y = arow[ka + 1];
        // unconditional loads from clamped column, then per-lane select (v_cndmask)
        float b0 = bcol[(size_t)ka * SPIX];
        float b1 = bcol[(size_t)(ka + 1) * SPIX];
        v2f bv;
        bv.x = tin ? b0 : 0.0f;
        bv.y = tin ? b1 : 0.0f;
        acc = __builtin_amdgcn_wmma_f32_16x16x4_f32(
            /*neg_a=*/false, a, /*neg_b=*/false, bv,
            /*c_mod=*/(short)0, acc, /*reuse_a=*/false, /*reuse_b=*/false);
    }
    // trace partial: sum_v T[m0+v+8*hi][t] * R[m0+v+8*hi][t]  (padded region is zero)
    float partial = 0.0f;
    const float* __restrict__ rcol = Rb + t;
#pragma unroll
    for (int v = 0; v < 8; ++v)
        partial += acc[v] * rcol[(size_t)(m0 + v + 8 * hi) * SPAD];
    for (int off = 16; off; off >>= 1) partial += __shfl_xor(partial, off, 32);
    if (lane == 0) atomicAdd(tsum, partial);
}

// ---------------- finalize ----------------
__global__ void k_final(const float* __restrict__ wss, const float* __restrict__ tsum,
                        const unsigned int* __restrict__ cnt, float* __restrict__ out) {
    float denom = (float)(*cnt) + 1e-16f;
    float norm  = sqrtf(*wss);
    out[0] = (2.0f / norm) * (*tsum) / (denom * denom) / (float)BATCH;
}

extern "C" void kernel_launch(void* const* d_in, const int* in_sizes, int n_in,
                              void* d_out, int out_size, void* d_ws, size_t ws_size,
                              hipStream_t stream) {
    const float* pred = (const float*)d_in[0];   // [8,21,512,512]
    const float* Wc   = (const float*)d_in[1];   // [8,1000,1000]
    const float* Lc   = (const float*)d_in[2];   // [8,1000,1000]
    const int*   seg  = (const int*)d_in[3];     // [8,512,512]

    float*        ws   = (float*)d_ws;
    float*        wss  = ws + 0;
    float*        tsum = ws + 1;
    unsigned int* cnt  = (unsigned int*)(ws + 2);
    float*        R    = ws + 16;                // [8][32][1008] f32, ~1.03 MB

    const int nz = 16 + BATCH * MPAD * SPAD;
    k_zero<<<(nz + 255) / 256, 256, 0, stream>>>(ws, nz);

    const unsigned int ntot = BATCH * CLASSES * PIX;     // divisible by 256
    k_segsum<<<ntot / 256, 256, 0, stream>>>(pred, seg, R);

    k_mask<<<256, 256, 0, stream>>>(seg, cnt);
    k_wnorm<<<1024, 256, 0, stream>>>(Wc, wss, BATCH * SPIX * SPIX);

    k_trace<<<BATCH * 2 * NTILES, 32, 0, stream>>>(Lc, R, tsum);

    k_final<<<1, 1, 0, stream>>>(wss, tsum, cnt, (float*)d_out);
}